// self_attention_fc_51161650430035
// MI455X (gfx1250) — compile-verified
//
#include <hip/hip_runtime.h>

typedef float v2f __attribute__((ext_vector_type(2)));
typedef float v8f __attribute__((ext_vector_type(8)));

#define LSTR 68            // LDS row stride in floats (64 + 4 pad)
#define NC   64            // channels
#define DHW  (64*64*64)    // per-(b,c) spatial extent

__device__ __forceinline__ float mish_f(float t) {
    // mish(t) = t * tanh(softplus(t)) = t * (p^2-1)/(p^2+1), p = 1+e^t
    float p  = 1.0f + __expf(fminf(t, 20.0f));
    float p2 = p * p;
    return t * __fdividef(p2 - 1.0f, p2 + 1.0f);
}

__global__ void __launch_bounds__(256)
fused_channel_attn_kernel(const float* __restrict__ x,
                          const float* __restrict__ y,
                          const float* __restrict__ wq,
                          const float* __restrict__ beta_p,
                          float* __restrict__ out)
{
    __shared__ float sMx[NC * LSTR];   // mish(x) row tile, later reused as attn_xy
    __shared__ float sMy[NC * LSTR];   // mish(y) row tile, later reused as attn_yx
    __shared__ float sPx[NC * LSTR];   // qx = Wq @ mish(x)
    __shared__ float sPy[NC * LSTR];   // qy = Wq @ mish(y)
    __shared__ float sE [NC * LSTR];   // energy

    const int tid  = threadIdx.x;
    const int lane = tid & 31;
    const int wid  = tid >> 5;
    const int m    = lane & 15;   // M/N index within fragment
    const int kh   = lane >> 4;   // K-half select (lanes 16-31 hold K+2)

    const int bid = blockIdx.x;          // 8192 = 2 * 64 * 64 rows
    const int b   = bid >> 12;
    const int dh  = bid & 4095;          // d*64 + h
    const size_t rowoff = (size_t)dh * 64;

    // ---------------- Phase 0: load + mish into LDS (float4, coalesced) ----
    {
        const float* xb = x + (size_t)b * NC * DHW + rowoff;
        const float* yb = y + (size_t)b * NC * DHW + rowoff;
        #pragma unroll
        for (int i = 0; i < 4; ++i) {
            int j = tid + 256 * i;             // float4 slot 0..1023
            int c = j >> 4;
            int w = (j & 15) * 4;
            float4 vx = *(const float4*)(xb + (size_t)c * DHW + w);
            float4 vy = *(const float4*)(yb + (size_t)c * DHW + w);
            float4 rx, ry;
            rx.x = mish_f(vx.x); rx.y = mish_f(vx.y);
            rx.z = mish_f(vx.z); rx.w = mish_f(vx.w);
            ry.x = mish_f(vy.x); ry.y = mish_f(vy.y);
            ry.z = mish_f(vy.z); ry.w = mish_f(vy.w);
            *(float4*)(sMx + c * LSTR + w) = rx;
            *(float4*)(sMy + c * LSTR + w) = ry;
        }
    }
    __syncthreads();

    // ---------------- Phase A: P = Wq @ M  (f32 WMMA, K=4 x 16 steps) ------
    {
        const int    sel = wid & 1;                  // 0 -> x path, 1 -> y path
        const int    o0  = (wid >> 1) * 16;          // output-channel strip
        const float* src = sel ? sMy : sMx;
        float*       dst = sel ? sPy : sPx;

        v2f aF[16];                                  // Wq[o0+m][k..k+1], K-half kh
        const float* wrow = wq + (o0 + m) * NC + 2 * kh;
        #pragma unroll
        for (int kk = 0; kk < 16; ++kk)
            aF[kk] = *(const v2f*)(wrow + 4 * kk);

        #pragma unroll
        for (int nt = 0; nt < 4; ++nt) {
            const int n0 = nt * 16;
            v8f acc = {0.f,0.f,0.f,0.f,0.f,0.f,0.f,0.f};
            #pragma unroll
            for (int kk = 0; kk < 16; ++kk) {
                const int kr = 4 * kk + 2 * kh;
                v2f bF;
                bF.x = src[(kr    ) * LSTR + n0 + m];
                bF.y = src[(kr + 1) * LSTR + n0 + m];
                acc = __builtin_amdgcn_wmma_f32_16x16x4_f32(
                        false, aF[kk], false, bF, (short)0, acc, false, false);
            }
            #pragma unroll
            for (int rr = 0; rr < 8; ++rr)
                dst[(o0 + rr + 8 * kh) * LSTR + n0 + m] = acc[rr];
        }
    }
    __syncthreads();

    // ---------------- Phase B: E = Px @ Py^T  (both frags contiguous f2) ---
    {
        #pragma unroll
        for (int t2 = 0; t2 < 2; ++t2) {
            const int t  = wid * 2 + t2;
            const int c0 = (t >> 2) * 16;
            const int e0 = (t & 3) * 16;
            v8f acc = {0.f,0.f,0.f,0.f,0.f,0.f,0.f,0.f};
            #pragma unroll
            for (int kk = 0; kk < 16; ++kk) {
                const int kr = 4 * kk + 2 * kh;
                v2f aF = *(const v2f*)(sPx + (c0 + m) * LSTR + kr);
                v2f bF = *(const v2f*)(sPy + (e0 + m) * LSTR + kr);
                acc = __builtin_amdgcn_wmma_f32_16x16x4_f32(
                        false, aF, false, bF, (short)0, acc, false, false);
            }
            #pragma unroll
            for (int rr = 0; rr < 8; ++rr)
                sE[(c0 + rr + 8 * kh) * LSTR + e0 + m] = acc[rr];
        }
    }
    __syncthreads();

    // ---------------- Phase C: dual softmax (rows -> sMx, cols -> sMy) -----
    if (tid < 128) {
        const int  c    = tid & 63;
        const bool colp = tid >= 64;                 // column softmax path
        float*     dst  = colp ? sMy : sMx;          // attn_yx : attn_xy
        float mx = -3.4e38f;
        for (int e = 0; e < 64; ++e) {
            float v = colp ? sE[e * LSTR + c] : sE[c * LSTR + e];
            mx = fmaxf(mx, v);
        }
        float s = 0.f;
        for (int e = 0; e < 64; ++e) {
            float v  = colp ? sE[e * LSTR + c] : sE[c * LSTR + e];
            float ex = __expf(v - mx);
            dst[c * LSTR + e] = ex;
            s += ex;
        }
        float inv = 1.0f / s;
        for (int e = 0; e < 64; ++e) dst[c * LSTR + e] *= inv;
    }
    __syncthreads();

    // ---------------- Phase D: out = beta * (attn @ P) + P, store ----------
    {
        const float  beta = beta_p[0];
        const int    sel  = wid & 1;                 // 0 -> out_x, 1 -> out_y
        const int    o0   = (wid >> 1) * 16;
        const float* Am   = sel ? sMy : sMx;
        const float* Pm   = sel ? sPy : sPx;
        float* ob = out + (size_t)(sel * 2 + b) * NC * DHW + rowoff;

        #pragma unroll
        for (int nt = 0; nt < 4; ++nt) {
            const int n0 = nt * 16;
            v8f acc = {0.f,0.f,0.f,0.f,0.f,0.f,0.f,0.f};
            #pragma unroll
            for (int kk = 0; kk < 16; ++kk) {
                const int kr = 4 * kk + 2 * kh;
                v2f aF = *(const v2f*)(Am + (o0 + m) * LSTR + kr);
                v2f bF;
                bF.x = Pm[(kr    ) * LSTR + n0 + m];
                bF.y = Pm[(kr + 1) * LSTR + n0 + m];
                acc = __builtin_amdgcn_wmma_f32_16x16x4_f32(
                        false, aF, false, bF, (short)0, acc, false, false);
            }
            #pragma unroll
            for (int rr = 0; rr < 8; ++rr) {
                const int row = o0 + rr + 8 * kh;
                const int cl  = n0 + m;
                float v = beta * acc[rr] + Pm[row * LSTR + cl];
                ob[(size_t)row * DHW + cl] = v;
            }
        }
    }
}

extern "C" void kernel_launch(void* const* d_in, const int* in_sizes, int n_in,
                              void* d_out, int out_size, void* d_ws, size_t ws_size,
                              hipStream_t stream) {
    const float* x    = (const float*)d_in[0];
    const float* y    = (const float*)d_in[1];
    const float* wq   = (const float*)d_in[2];
    const float* beta = (const float*)d_in[3];
    float* out        = (float*)d_out;
    (void)in_sizes; (void)n_in; (void)out_size; (void)d_ws; (void)ws_size;

    dim3 grid(8192);   // 2 * 64 * 64 (b, d, h) rows
    dim3 block(256);   // 8 wave32
    hipLaunchKernelGGL(fused_channel_attn_kernel, grid, block, 0, stream,
                       x, y, wq, beta, out);
}